// GNN_65532611002729
// MI455X (gfx1250) — compile-verified
//
#include <hip/hip_runtime.h>
#include <cstdint>
#include <cstddef>

typedef __attribute__((ext_vector_type(2))) float v2f;
typedef __attribute__((ext_vector_type(8))) float v8f;

// ---------------------------------------------------------------------------
// Utility kernels
// ---------------------------------------------------------------------------
__global__ void fill_zero_k(float* __restrict__ p, size_t n) {
    size_t i = (size_t)blockIdx.x * blockDim.x + threadIdx.x;
    if (i < n) p[i] = 0.0f;
}

__global__ void reduce_partial_k(const float* __restrict__ v, float* __restrict__ out, int n) {
    __shared__ float sm[256];
    float s = 0.0f;
    for (int i = blockIdx.x * 256 + threadIdx.x; i < n; i += 256 * 256)
        s += v[i];
    sm[threadIdx.x] = s;
    __syncthreads();
    for (int off = 128; off > 0; off >>= 1) {
        if (threadIdx.x < off) sm[threadIdx.x] += sm[threadIdx.x + off];
        __syncthreads();
    }
    if (threadIdx.x == 0) out[blockIdx.x] = sm[0];
}

__global__ void reduce_final_k(const float* __restrict__ part, float* __restrict__ out) {
    __shared__ float sm[256];
    sm[threadIdx.x] = part[threadIdx.x];
    __syncthreads();
    for (int off = 128; off > 0; off >>= 1) {
        if (threadIdx.x < off) sm[threadIdx.x] += sm[threadIdx.x + off];
        __syncthreads();
    }
    if (threadIdx.x == 0) out[0] = sm[0];
}

// ---------------------------------------------------------------------------
// GEMM: out = A[M x 128] @ [Wl | Wr] ([128 x 128] each), fp32 WMMA 16x16x4.
// One wave per 16x16 output tile. M must be a multiple of 16 (50000 = 3125*16).
//
// A-operand layout (16x4 f32): lane 0-15 -> row M=lane, K = {0,1}; lanes
// 16-31 -> K = {2,3}. B-operand (4x16, column-major striping): lane 0-15 ->
// col N=lane, K={0,1}; lanes 16-31 -> K={2,3}. C/D: VGPR r holds row
// M = r + 8*(lane>>4), col N = lane&15.
// ---------------------------------------------------------------------------
__global__ void gemm_xlxr_k(const float* __restrict__ A,
                            const float* __restrict__ Wl,
                            const float* __restrict__ Wr,
                            float* __restrict__ xl,
                            float* __restrict__ xr,
                            int M) {
    const int wave = (int)((blockIdx.x * blockDim.x + threadIdx.x) >> 5);
    const int lane = threadIdx.x & 31;
    const int numMt = M >> 4;
    const int mt = wave >> 4;          // 16 n-tiles (256 output cols)
    const int nt = wave & 15;
    if (mt >= numMt) return;           // wave-uniform

    const float* __restrict__ W  = (nt < 8) ? Wl : Wr;
    float* __restrict__ out      = (nt < 8) ? xl : xr;
    const int n0   = (nt & 7) * 16;
    const int m0   = mt * 16;
    const int mrow = m0 + (lane & 15);
    const int ncol = n0 + (lane & 15);
    const int koff = (lane >> 4) * 2;

    v8f c = {0.f, 0.f, 0.f, 0.f, 0.f, 0.f, 0.f, 0.f};
    const float* __restrict__ arow = A + (size_t)mrow * 128;

#pragma unroll 8
    for (int k0 = 0; k0 < 128; k0 += 4) {
        v2f a, b;
        a.x = arow[k0 + koff + 0];
        a.y = arow[k0 + koff + 1];
        b.x = W[(size_t)(k0 + koff + 0) * 128 + ncol];
        b.y = W[(size_t)(k0 + koff + 1) * 128 + ncol];
        c = __builtin_amdgcn_wmma_f32_16x16x4_f32(
                /*neg_a=*/false, a, /*neg_b=*/false, b,
                /*c_mod=*/(short)0, c, /*reuse_a=*/false, /*reuse_b=*/false);
    }

    const int obase = m0 + 8 * (lane >> 4);
#pragma unroll
    for (int r = 0; r < 8; ++r)
        out[(size_t)(obase + r) * 128 + ncol] = c[r];
}

// ---------------------------------------------------------------------------
// Monotone float<->uint mapping for atomicMax-based segment max.
// ---------------------------------------------------------------------------
__device__ __forceinline__ unsigned fkey(float f) {
    unsigned b = __float_as_uint(f);
    return b ^ (unsigned)(((int)b >> 31) | 0x80000000);
}
__device__ __forceinline__ float fdecode(unsigned u) {
    return (u & 0x80000000u) ? __uint_as_float(u ^ 0x80000000u)
                             : __uint_as_float(~u);
}

// ---------------------------------------------------------------------------
// Pass A: per-edge attention logits + segment max. One wave per edge,
// 4 contiguous channels per lane (float4), heads reduced over 8-lane groups.
// ---------------------------------------------------------------------------
__global__ void edge_logits_k(const int* __restrict__ ei,
                              const float* __restrict__ ew,
                              const float* __restrict__ xl,
                              const float* __restrict__ xr,
                              const float* __restrict__ We,
                              const float* __restrict__ att,
                              const float* __restrict__ sumw,
                              float* __restrict__ logits,
                              unsigned* __restrict__ mkey,
                              int E, int Nn) {
    const int e    = (int)((blockIdx.x * blockDim.x + threadIdx.x) >> 5);
    const int lane = threadIdx.x & 31;
    const int Etot = E + Nn;
    if (e >= Etot) return;              // wave-uniform

    int src, dst; float w;
    if (e < E) { src = ei[e]; dst = ei[E + e]; w = ew[e]; }
    else       { src = e - E; dst = src; w = sumw[0] * (1.0f / (float)E); }

    const int c4 = lane * 4;
    const float4 a  = *(const float4*)(xl  + (size_t)src * 128 + c4);
    const float4 r  = *(const float4*)(xr  + (size_t)dst * 128 + c4);
    const float4 wv = *(const float4*)(We  + c4);
    const float4 at = *(const float4*)(att + c4);

    float z, partial = 0.0f;
    z = a.x + r.x + w * wv.x; z = (z > 0.f) ? z : 0.2f * z; partial += z * at.x;
    z = a.y + r.y + w * wv.y; z = (z > 0.f) ? z : 0.2f * z; partial += z * at.y;
    z = a.z + r.z + w * wv.z; z = (z > 0.f) ? z : 0.2f * z; partial += z * at.z;
    z = a.w + r.w + w * wv.w; z = (z > 0.f) ? z : 0.2f * z; partial += z * at.w;

    partial += __shfl_xor(partial, 1, 32);
    partial += __shfl_xor(partial, 2, 32);
    partial += __shfl_xor(partial, 4, 32);

    if ((lane & 7) == 0) {
        const int h = lane >> 3;
        logits[(size_t)e * 4 + h] = partial;
        atomicMax(&mkey[(size_t)dst * 4 + h], fkey(partial));
    }
}

// ---------------------------------------------------------------------------
// Pass B: p = exp(logit - m[dst]); denom[dst] += p. One thread per (edge,head).
// unsafeAtomicAdd -> native GLOBAL_ATOMIC_ADD_F32 (no CAS loop).
// ---------------------------------------------------------------------------
__global__ void edge_softmax_k(const int* __restrict__ ei,
                               float* __restrict__ logits,
                               const unsigned* __restrict__ mkey,
                               float* __restrict__ denom,
                               int E, int Nn) {
    const int Etot = E + Nn;
    const size_t idx = (size_t)blockIdx.x * blockDim.x + threadIdx.x;
    if (idx >= (size_t)Etot * 4) return;
    const int e = (int)(idx >> 2);
    const int h = (int)(idx & 3);
    const int dst = (e < E) ? ei[E + e] : (e - E);
    const float m = fdecode(mkey[(size_t)dst * 4 + h]);
    const float p = expf(logits[idx] - m);
    logits[idx] = p;
    unsafeAtomicAdd(&denom[(size_t)dst * 4 + h], p);
}

// ---------------------------------------------------------------------------
// Pass C: acc[dst] += (p / (denom[dst]+1e-16)) * xl[src]. One wave per edge.
// unsafeAtomicAdd -> native GLOBAL_ATOMIC_ADD_F32 on the hot scatter path.
// ---------------------------------------------------------------------------
__global__ void edge_aggregate_k(const int* __restrict__ ei,
                                 const float* __restrict__ pbuf,
                                 const float* __restrict__ denom,
                                 const float* __restrict__ xl,
                                 float* __restrict__ acc,
                                 int E, int Nn) {
    const int e    = (int)((blockIdx.x * blockDim.x + threadIdx.x) >> 5);
    const int lane = threadIdx.x & 31;
    const int Etot = E + Nn;
    if (e >= Etot) return;              // wave-uniform

    int src, dst;
    if (e < E) { src = ei[e]; dst = ei[E + e]; }
    else       { src = e - E; dst = src; }

    const int h = lane >> 3;
    const float p = pbuf[(size_t)e * 4 + h];
    const float d = denom[(size_t)dst * 4 + h];
    const float alpha = p / (d + 1e-16f);

    const int c4 = lane * 4;
    const float4 a = *(const float4*)(xl + (size_t)src * 128 + c4);
    float* __restrict__ o = acc + (size_t)dst * 128 + c4;
    unsafeAtomicAdd(o + 0, alpha * a.x);
    unsafeAtomicAdd(o + 1, alpha * a.y);
    unsafeAtomicAdd(o + 2, alpha * a.z);
    unsafeAtomicAdd(o + 3, alpha * a.w);
}

// ---------------------------------------------------------------------------
// h = elu(acc + b1)     (layer-1 epilogue, 128 channels)
// ---------------------------------------------------------------------------
__global__ void bias_elu_k(const float* __restrict__ acc,
                           const float* __restrict__ b,
                           float* __restrict__ h, size_t n) {
    size_t i = (size_t)blockIdx.x * blockDim.x + threadIdx.x;
    if (i >= n) return;
    const float v = acc[i] + b[i & 127];
    h[i] = (v > 0.f) ? v : (expf(v) - 1.0f);
}

// ---------------------------------------------------------------------------
// out[i,c] = mean_h(acc[i, h*32+c]) + b2[c]   (layer-2 epilogue)
// ---------------------------------------------------------------------------
__global__ void head_mean_k(const float* __restrict__ acc,
                            const float* __restrict__ b2,
                            float* __restrict__ out, int Nn) {
    const size_t idx = (size_t)blockIdx.x * blockDim.x + threadIdx.x;
    if (idx >= (size_t)Nn * 32) return;
    const int i = (int)(idx >> 5);
    const int c = (int)(idx & 31);
    const float* row = acc + (size_t)i * 128 + c;
    const float s = row[0] + row[32] + row[64] + row[96];
    out[idx] = 0.25f * s + b2[c];
}

// ---------------------------------------------------------------------------
// Host launcher
// ---------------------------------------------------------------------------
extern "C" void kernel_launch(void* const* d_in, const int* in_sizes, int n_in,
                              void* d_out, int out_size, void* d_ws, size_t ws_size,
                              hipStream_t stream) {
    const float* x    = (const float*)d_in[0];
    const int*   ei   = (const int*)  d_in[1];
    const float* ew   = (const float*)d_in[2];
    const float* Wl1  = (const float*)d_in[3];
    const float* Wr1  = (const float*)d_in[4];
    const float* We1  = (const float*)d_in[5];
    const float* att1 = (const float*)d_in[6];
    const float* b1   = (const float*)d_in[7];
    const float* Wl2  = (const float*)d_in[8];
    const float* Wr2  = (const float*)d_in[9];
    const float* We2  = (const float*)d_in[10];
    const float* att2 = (const float*)d_in[11];
    const float* b2   = (const float*)d_in[12];

    const int N    = in_sizes[0] / 128;   // 50000
    const int E    = in_sizes[2];         // 1,600,000
    const int Etot = E + N;
    const size_t nf = (size_t)N * 128;

    // workspace carve-up (all chunks 256B-aligned by construction)
    char* p = (char*)d_ws;
    float*    xl     = (float*)p;    p += nf * 4;
    float*    xr     = (float*)p;    p += nf * 4;
    float*    hbuf   = (float*)p;    p += nf * 4;
    float*    acc    = (float*)p;    p += nf * 4;
    float*    logits = (float*)p;    p += (size_t)Etot * 4 * 4;
    unsigned* mkey   = (unsigned*)p; p += (size_t)N * 4 * 4;
    float*    denom  = (float*)p;    p += (size_t)N * 4 * 4;
    float*    parts  = (float*)p;    p += 256 * 4;
    float*    sumw   = (float*)p;    p += 256;      // scalar, padded

    const int blk_gemm = ((N / 16) * 16 * 32 + 255) / 256;        // waves = tiles
    const int blk_ew   = (int)(((size_t)Etot * 32 + 255) / 256);  // wave/edge
    const int blk_e4   = (int)(((size_t)Etot * 4 + 255) / 256);
    const int blk_nf   = (int)((nf + 255) / 256);
    const int blk_n4   = (N * 4 + 255) / 256;
    const int blk_n32  = (N * 32 + 255) / 256;

    // mean(edge_weight), deterministic two-stage reduction
    reduce_partial_k<<<256, 256, 0, stream>>>(ew, parts, E);
    reduce_final_k<<<1, 256, 0, stream>>>(parts, sumw);

    // ---------------- layer 1 ----------------
    gemm_xlxr_k<<<blk_gemm, 256, 0, stream>>>(x, Wl1, Wr1, xl, xr, N);
    fill_zero_k<<<blk_nf, 256, 0, stream>>>(acc, nf);
    fill_zero_k<<<blk_n4, 256, 0, stream>>>((float*)mkey, (size_t)N * 4);
    fill_zero_k<<<blk_n4, 256, 0, stream>>>(denom, (size_t)N * 4);
    edge_logits_k<<<blk_ew, 256, 0, stream>>>(ei, ew, xl, xr, We1, att1, sumw,
                                              logits, mkey, E, N);
    edge_softmax_k<<<blk_e4, 256, 0, stream>>>(ei, logits, mkey, denom, E, N);
    edge_aggregate_k<<<blk_ew, 256, 0, stream>>>(ei, logits, denom, xl, acc, E, N);
    bias_elu_k<<<blk_nf, 256, 0, stream>>>(acc, b1, hbuf, nf);

    // ---------------- layer 2 ----------------
    gemm_xlxr_k<<<blk_gemm, 256, 0, stream>>>(hbuf, Wl2, Wr2, xl, xr, N);
    fill_zero_k<<<blk_nf, 256, 0, stream>>>(acc, nf);
    fill_zero_k<<<blk_n4, 256, 0, stream>>>((float*)mkey, (size_t)N * 4);
    fill_zero_k<<<blk_n4, 256, 0, stream>>>(denom, (size_t)N * 4);
    edge_logits_k<<<blk_ew, 256, 0, stream>>>(ei, ew, xl, xr, We2, att2, sumw,
                                              logits, mkey, E, N);
    edge_softmax_k<<<blk_e4, 256, 0, stream>>>(ei, logits, mkey, denom, E, N);
    edge_aggregate_k<<<blk_ew, 256, 0, stream>>>(ei, logits, denom, xl, acc, E, N);
    head_mean_k<<<blk_n32, 256, 0, stream>>>(acc, b2, (float*)d_out, N);
}